// MBGCN_15255723836096
// MI455X (gfx1250) — compile-verified
//
#include <hip/hip_runtime.h>

typedef __attribute__((ext_vector_type(2))) float v2f;
typedef __attribute__((ext_vector_type(8))) float v8f;

constexpr int U_ = 100000, I_ = 50000, D_ = 64, L_ = 2, T_ = 3;
constexpr int E_UI_ = 1000000, E_II_ = 800000, B_ = 8192;
constexpr float LAMB_ = 0.5f;

// ---------------------------------------------------------------- degrees
__global__ void k_count(const int* __restrict__ ids, float* __restrict__ deg, int nE) {
  int e = blockIdx.x * blockDim.x + threadIdx.x;
  if (e < nE) atomicAdd(&deg[ids[e]], 1.0f);
}

__global__ void k_clamp1(float* __restrict__ deg, int n) {
  int i = blockIdx.x * blockDim.x + threadIdx.x;
  if (i < n) deg[i] = fmaxf(deg[i], 1.0f);
}

// ---------------------------------------------------------------- scatter
// agg[seg[e]] += coef * emb[gat[e]] / deg[seg[e]]   (one thread per edge-dim)
__global__ void k_scatter(const int* __restrict__ seg, const int* __restrict__ gat,
                          const float* __restrict__ emb, const float* __restrict__ deg,
                          const float* __restrict__ alpha, int t,
                          float* __restrict__ agg, int nE) {
  unsigned tid = blockIdx.x * blockDim.x + threadIdx.x;
  int e = (int)(tid >> 6);
  int d = (int)(tid & 63u);
  if (e >= nE) return;
  if (d == 0) {  // run ahead on the index streams (global_prefetch_b8)
    __builtin_prefetch(seg + e + 2048, 0, 0);
    __builtin_prefetch(gat + e + 2048, 0, 0);
  }
  int s = seg[e], g = gat[e];
  float coef = 1.0f;
  if (alpha) coef = alpha[t] / (alpha[0] + alpha[1] + alpha[2]);
  float v = coef * emb[(size_t)g * D_ + d] / deg[s];
  atomicAdd(&agg[(size_t)s * D_ + d], v);
}

// ---------------------------------------------------------------- WMMA GEMM + LayerNorm
// Y = LayerNorm(X @ W^T) ; X:[N,64], W:[64,64]. Block = 16 rows, 4 waves (one
// 16x16 column tile per wave), K=64 via 16x V_WMMA_F32_16X16X4_F32 steps.
__global__ void k_gemm_ln(const float* __restrict__ X, const float* __restrict__ W,
                          const float* __restrict__ gamma, const float* __restrict__ beta,
                          float* __restrict__ Y, int N) {
  __shared__ float tile[16][64];
  __shared__ float s_mean[16], s_inv[16];
  const int lane = threadIdx.x & 31;
  const int wave = threadIdx.x >> 5;          // 0..3
  const int row0 = blockIdx.x * 16;
  const int r    = lane & 15;                 // row within tile (A) / col within tile (B)
  const int koff = (lane >> 4) << 1;          // 0 (lanes 0-15) or 2 (lanes 16-31)
  const int col0 = wave << 4;
  v8f c = {};
#pragma unroll
  for (int k0 = 0; k0 < 64; k0 += 4) {
    v2f a, b;
    const float* xp = X + (size_t)(row0 + r) * D_ + k0 + koff;
    a.x = xp[0]; a.y = xp[1];                 // A[M=r][K=k0+koff(+1)]
    const float* wp = W + (size_t)(col0 + r) * D_ + k0 + koff;  // B[k][n] = W[n][k]
    b.x = wp[0]; b.y = wp[1];
    c = __builtin_amdgcn_wmma_f32_16x16x4_f32(false, a, false, b, (short)0, c, false, false);
  }
  const int rbase = (lane >> 4) << 3;         // C/D: VGPR v -> rows v and v+8
#pragma unroll
  for (int v = 0; v < 8; ++v) tile[rbase + v][col0 + r] = c[v];
  __syncthreads();
  if (threadIdx.x < 16) {
    int rr = threadIdx.x;
    float m = 0.f;
#pragma unroll
    for (int j = 0; j < 64; ++j) m += tile[rr][j];
    m *= (1.0f / 64.0f);
    float var = 0.f;
#pragma unroll
    for (int j = 0; j < 64; ++j) { float dd = tile[rr][j] - m; var += dd * dd; }
    var *= (1.0f / 64.0f);
    s_mean[rr] = m;
    s_inv[rr]  = rsqrtf(var + 1e-5f);
  }
  __syncthreads();
  for (int i = threadIdx.x; i < 16 * 64; i += blockDim.x) {
    int rr = i >> 6, cc = i & 63;
    Y[(size_t)(row0 + rr) * D_ + cc] =
        (tile[rr][cc] - s_mean[rr]) * s_inv[rr] * gamma[cc] + beta[cc];
  }
}

// ---------------------------------------------------------------- WMMA GEMM + ReLU
// Y = relu(X' @ Wmat) with X' rows optionally gathered via rowidx.
// transB=1: Y = relu(X' @ W^T) ; transB=0: Y = relu(X' @ W)
__global__ void k_gemm_relu(const float* __restrict__ X, const int* __restrict__ rowidx,
                            const float* __restrict__ W, float* __restrict__ Y,
                            int N, int transB) {
  const int lane = threadIdx.x & 31;
  const int wave = threadIdx.x >> 5;
  const int row0 = blockIdx.x * 16;
  const int r    = lane & 15;
  const int koff = (lane >> 4) << 1;
  const int col0 = wave << 4;
  int ar = row0 + r;
  if (rowidx) ar = rowidx[ar];
  const float* xbase = X + (size_t)ar * D_;
  v8f c = {};
#pragma unroll
  for (int k0 = 0; k0 < 64; k0 += 4) {
    v2f a, b;
    a.x = xbase[k0 + koff]; a.y = xbase[k0 + koff + 1];
    if (transB) {
      const float* wp = W + (size_t)(col0 + r) * D_ + k0 + koff;   // B[k][n] = W[n][k]
      b.x = wp[0]; b.y = wp[1];
    } else {
      b.x = W[(size_t)(k0 + koff) * D_ + col0 + r];                // B[k][n] = W[k][n]
      b.y = W[(size_t)(k0 + koff + 1) * D_ + col0 + r];
    }
    c = __builtin_amdgcn_wmma_f32_16x16x4_f32(false, a, false, b, (short)0, c, false, false);
  }
  const int rbase = (lane >> 4) << 3;
#pragma unroll
  for (int v = 0; v < 8; ++v) {
    float val = c[v];
    Y[(size_t)(row0 + rbase + v) * D_ + col0 + r] = val > 0.f ? val : 0.f;
  }
}

// ---------------------------------------------------------------- scoring dots
__global__ void k_user_cf(const float* __restrict__ u, const float* __restrict__ it,
                          const int* __restrict__ uidx, const int* __restrict__ iidx,
                          float* __restrict__ out, int Bn, float scale) {
  int w    = (int)((blockIdx.x * blockDim.x + threadIdx.x) >> 5);
  int lane = threadIdx.x & 31;
  if (w >= Bn) return;
  const float* up = u  + (size_t)uidx[w] * D_;
  const float* ip = it + (size_t)iidx[w] * D_;
  float acc = up[lane] * ip[lane] + up[lane + 32] * ip[lane + 32];
  for (int off = 16; off > 0; off >>= 1) acc += __shfl_xor(acc, off, 32);
  if (lane == 0) out[w] = scale * acc;
}

__global__ void k_item_cf(const float* __restrict__ tb, const float* __restrict__ s,
                          const int* __restrict__ iidx, float* __restrict__ out,
                          int Bn, float scale) {
  int w    = (int)((blockIdx.x * blockDim.x + threadIdx.x) >> 5);
  int lane = threadIdx.x & 31;
  if (w >= Bn) return;
  const float* tp = tb + (size_t)w * D_;
  const float* sp = s  + (size_t)iidx[w] * D_;
  float acc = tp[lane] * sp[lane] + tp[lane + 32] * sp[lane + 32];
  for (int off = 16; off > 0; off >>= 1) acc += __shfl_xor(acc, off, 32);
  if (lane == 0) out[w] += scale * acc;
}

// ---------------------------------------------------------------- host orchestration
extern "C" void kernel_launch(void* const* d_in, const int* in_sizes, int n_in,
                              void* d_out, int out_size, void* d_ws, size_t ws_size,
                              hipStream_t stream) {
  const int*   user_idx   = (const int*)d_in[0];
  const int*   item_idx   = (const int*)d_in[1];
  const int*   ui_src     = (const int*)d_in[2];   // [T,E_UI]
  const int*   ui_dst     = (const int*)d_in[3];
  const int*   ii_src     = (const int*)d_in[4];   // [T,E_II]
  const int*   ii_dst     = (const int*)d_in[5];
  const float* user_emb   = (const float*)d_in[6];
  const float* item_emb   = (const float*)d_in[7];
  const float* s_item_emb = (const float*)d_in[8]; // [T,I,D]
  const float* alpha      = (const float*)d_in[9];
  const float* W_ui       = (const float*)d_in[10]; // [L,D,D]
  const float* W_ii       = (const float*)d_in[11]; // [L,T,D,D]
  const float* M_t        = (const float*)d_in[12]; // [T,D,D]
  const float* ln_scale   = (const float*)d_in[13]; // [L,D]
  const float* ln_bias    = (const float*)d_in[14];
  float* out = (float*)d_out;
  float* ws  = (float*)d_ws;

  // workspace layout (floats)
  size_t o = 0;
  float* deg_u = ws + o; o += (size_t)T_ * U_;
  float* deg_i = ws + o; o += (size_t)T_ * I_;
  float* deg_s = ws + o; o += (size_t)T_ * I_;
  float* ubuf[2]; ubuf[0] = ws + o; o += (size_t)U_ * D_; ubuf[1] = ws + o; o += (size_t)U_ * D_;
  float* ibuf[2]; ibuf[0] = ws + o; o += (size_t)I_ * D_; ibuf[1] = ws + o; o += (size_t)I_ * D_;
  float* sbuf[2]; sbuf[0] = ws + o; o += (size_t)T_ * I_ * D_; sbuf[1] = ws + o; o += (size_t)T_ * I_ * D_;
  float* agg_u = ws + o; o += (size_t)U_ * D_;
  float* agg_i = ws + o; o += (size_t)I_ * D_;
  float* agg_s = ws + o; o += (size_t)T_ * I_ * D_;
  float* um    = ws + o; o += (size_t)U_ * D_;
  float* tb    = ws + o; o += (size_t)B_ * D_;

  const int TPB = 256;
  auto cdiv = [](size_t a, size_t b) { return (int)((a + b - 1) / b); };

  // ---- degrees (computed once)
  hipMemsetAsync(deg_u, 0, (size_t)T_ * U_ * 4, stream);
  hipMemsetAsync(deg_i, 0, (size_t)T_ * I_ * 4, stream);
  hipMemsetAsync(deg_s, 0, (size_t)T_ * I_ * 4, stream);
  for (int t = 0; t < T_; ++t) {
    k_count<<<cdiv(E_UI_, TPB), TPB, 0, stream>>>(ui_src + (size_t)t * E_UI_, deg_u + (size_t)t * U_, E_UI_);
    k_count<<<cdiv(E_UI_, TPB), TPB, 0, stream>>>(ui_dst + (size_t)t * E_UI_, deg_i + (size_t)t * I_, E_UI_);
    k_count<<<cdiv(E_II_, TPB), TPB, 0, stream>>>(ii_dst + (size_t)t * E_II_, deg_s + (size_t)t * I_, E_II_);
  }
  k_clamp1<<<cdiv((size_t)T_ * U_, TPB), TPB, 0, stream>>>(deg_u, T_ * U_);
  k_clamp1<<<cdiv((size_t)T_ * I_, TPB), TPB, 0, stream>>>(deg_i, T_ * I_);
  k_clamp1<<<cdiv((size_t)T_ * I_, TPB), TPB, 0, stream>>>(deg_s, T_ * I_);

  // ---- seed ping-pong buffers
  hipMemcpyAsync(ubuf[0], user_emb,   (size_t)U_ * D_ * 4,        hipMemcpyDeviceToDevice, stream);
  hipMemcpyAsync(ibuf[0], item_emb,   (size_t)I_ * D_ * 4,        hipMemcpyDeviceToDevice, stream);
  hipMemcpyAsync(sbuf[0], s_item_emb, (size_t)T_ * I_ * D_ * 4,   hipMemcpyDeviceToDevice, stream);

  int cur = 0;
  for (int l = 0; l < L_; ++l) {
    int nxt = cur ^ 1;
    hipMemsetAsync(agg_u, 0, (size_t)U_ * D_ * 4, stream);
    hipMemsetAsync(agg_i, 0, (size_t)I_ * D_ * 4, stream);
    hipMemsetAsync(agg_s, 0, (size_t)T_ * I_ * D_ * 4, stream);
    for (int t = 0; t < T_; ++t) {
      // u_agg += a[t] * segsum(it[ui_dst], ui_src) / deg_u[t]
      k_scatter<<<cdiv((size_t)E_UI_ * D_, TPB), TPB, 0, stream>>>(
          ui_src + (size_t)t * E_UI_, ui_dst + (size_t)t * E_UI_,
          ibuf[cur], deg_u + (size_t)t * U_, alpha, t, agg_u, E_UI_);
      // i_agg += segsum(u[ui_src], ui_dst) / deg_i[t]
      k_scatter<<<cdiv((size_t)E_UI_ * D_, TPB), TPB, 0, stream>>>(
          ui_dst + (size_t)t * E_UI_, ui_src + (size_t)t * E_UI_,
          ubuf[cur], deg_i + (size_t)t * I_, nullptr, t, agg_i, E_UI_);
      // s_agg[t] = segsum(s[t][ii_src], ii_dst) / deg_s[t]
      k_scatter<<<cdiv((size_t)E_II_ * D_, TPB), TPB, 0, stream>>>(
          ii_dst + (size_t)t * E_II_, ii_src + (size_t)t * E_II_,
          sbuf[cur] + (size_t)t * I_ * D_, deg_s + (size_t)t * I_, nullptr, t,
          agg_s + (size_t)t * I_ * D_, E_II_);
    }
    // u/i projections + LayerNorm (shared W_ui[l])
    k_gemm_ln<<<U_ / 16, 128, 0, stream>>>(agg_u, W_ui + (size_t)l * D_ * D_,
                                           ln_scale + (size_t)l * D_, ln_bias + (size_t)l * D_,
                                           ubuf[nxt], U_);
    k_gemm_ln<<<I_ / 16, 128, 0, stream>>>(agg_i, W_ui + (size_t)l * D_ * D_,
                                           ln_scale + (size_t)l * D_, ln_bias + (size_t)l * D_,
                                           ibuf[nxt], I_);
    // per-behavior item projections + ReLU (X @ W_ii[l,t]^T)
    for (int t = 0; t < T_; ++t) {
      k_gemm_relu<<<I_ / 16, 128, 0, stream>>>(
          agg_s + (size_t)t * I_ * D_, nullptr,
          W_ii + ((size_t)l * T_ + t) * D_ * D_,
          sbuf[nxt] + (size_t)t * I_ * D_, I_, /*transB=*/1);
    }
    cur = nxt;
  }

  // ---- user-based CF: out = LAMB * dot(u[user_idx], it[item_idx])
  k_user_cf<<<cdiv((size_t)B_ * 32, TPB), TPB, 0, stream>>>(
      ubuf[cur], ibuf[cur], user_idx, item_idx, out, B_, LAMB_);

  // ---- item-based CF
  for (int t = 0; t < T_; ++t) {
    hipMemsetAsync(um, 0, (size_t)U_ * D_ * 4, stream);
    // um = segsum(s[t][ui_dst], ui_src) / deg_u[t]
    k_scatter<<<cdiv((size_t)E_UI_ * D_, TPB), TPB, 0, stream>>>(
        ui_src + (size_t)t * E_UI_, ui_dst + (size_t)t * E_UI_,
        sbuf[cur] + (size_t)t * I_ * D_, deg_u + (size_t)t * U_, nullptr, t, um, E_UI_);
    // tb = relu(um[user_idx] @ M_t[t])   (no transpose)
    k_gemm_relu<<<B_ / 16, 128, 0, stream>>>(um, user_idx,
                                             M_t + (size_t)t * D_ * D_, tb, B_, /*transB=*/0);
    // out += (1-LAMB) * dot(tb, s[t][item_idx])
    k_item_cf<<<cdiv((size_t)B_ * 32, TPB), TPB, 0, stream>>>(
        tb, sbuf[cur] + (size_t)t * I_ * D_, item_idx, out, B_, 1.0f - LAMB_);
  }
}